// SoapBubble_13022340841699
// MI455X (gfx1250) — compile-verified
//
#include <hip/hip_runtime.h>
#include <stdint.h>

// SoapBubble vertex dynamics, MI455X (gfx1250, wave32).
// x lives in d_out (2M x 3 f32). F (2M x 3 f32) + vol accumulator live in d_ws.
// Force pass streams the face-index array into LDS with gfx1250 async
// global->LDS copies (ASYNCcnt), double-buffered against compute.

namespace {
constexpr int   kVerts = 2000000;
constexpr int   kFaces = 4000000;
constexpr int   kIters = 10;
constexpr float kDt    = 1e-8f;
constexpr float kSigma = 1.0f;
constexpr float kBulk  = 2500.0f;
constexpr float kEps   = 1e-12f;
// TARGET_CELL_VOLUME * exp(PRESSURE0 / BULK_MODULUS) = exp(0.04)
constexpr float kVt    = 1.0408107741923882f;
constexpr int   kBlock = 256;                 // 8 wave32 waves per block
constexpr int   kTileInts  = 3 * kBlock;      // 768 ints = 3072 B per face tile
constexpr int   kTileBytes = kTileInts * 4;
}

__device__ __forceinline__ void fatomic_add(float* p, float v) {
  // Lowers to global_atomic_add_f32 (no CAS loop) on gfx1250.
  unsafeAtomicAdd(p, v);
}

__device__ __forceinline__ void load3(const float* __restrict__ x, int i,
                                      float& a, float& b, float& c) {
  const float* p = x + 3 * (long)i;
  a = p[0]; b = p[1]; c = p[2];
}

// ---- gfx1250 async global->LDS copy (ASYNCcnt-tracked) ------------------
// lds_off: byte offset into this workgroup's LDS; goff: byte offset from base.
__device__ __forceinline__ void async_g2l_b32(uint32_t lds_off, uint32_t goff,
                                              const void* base) {
  asm volatile("global_load_async_to_lds_b32 %0, %1, %2"
               :: "v"(lds_off), "v"(goff), "s"(base)
               : "memory");
}

template <int N>
__device__ __forceinline__ void wait_asynccnt() {
#if __has_builtin(__builtin_amdgcn_s_wait_asynccnt)
  __builtin_amdgcn_s_wait_asynccnt(N);
#else
  if constexpr (N == 0) asm volatile("s_wait_asynccnt 0x0" ::: "memory");
  else                  asm volatile("s_wait_asynccnt 0x3" ::: "memory");
#endif
}

// issue one 3072B face tile: each of 256 lanes copies words tid, tid+256, tid+512
__device__ __forceinline__ void issue_tile(uint32_t lds_off0, const int* faces,
                                           int tile, int tid) {
  uint32_t goff = (uint32_t)tile * (uint32_t)kTileBytes + (uint32_t)tid * 4u;
  async_g2l_b32(lds_off0,         goff,         faces);
  async_g2l_b32(lds_off0 + 1024u, goff + 1024u, faces);
  async_g2l_b32(lds_off0 + 2048u, goff + 2048u, faces);
}

// ---------------- init: x = vertices, F = 0, vol = 0 --------------------
__global__ __launch_bounds__(kBlock)
void init_kernel(const float* __restrict__ verts, float* __restrict__ x,
                 float* __restrict__ F, float* __restrict__ vol) {
  int i = blockIdx.x * blockDim.x + threadIdx.x;
  if (i < 3 * kVerts) {
    x[i] = verts[i];
    F[i] = 0.0f;
  }
  if (i == 0) *vol = 0.0f;
}

// ---------------- pass 1: vol_raw = sum over faces of dot(v0, v1 x v2) --
__global__ __launch_bounds__(kBlock)
void vol_kernel(const float* __restrict__ x, const int* __restrict__ faces,
                float* __restrict__ vol) {
  __shared__ float red[kBlock / 32];
  float acc = 0.0f;
  const int stride = gridDim.x * blockDim.x;
  for (int f = blockIdx.x * blockDim.x + threadIdx.x; f < kFaces; f += stride) {
    // gfx1250 speculative prefetch of the streaming index array (far ahead).
    __builtin_prefetch(faces + 3 * (long)f + 3 * (long)stride, 0, 0);
    int i0 = faces[3 * (long)f + 0];
    int i1 = faces[3 * (long)f + 1];
    int i2 = faces[3 * (long)f + 2];
    float ax, ay, az, bx, by, bz, cx, cy, cz;
    load3(x, i0, ax, ay, az);
    load3(x, i1, bx, by, bz);
    load3(x, i2, cx, cy, cz);
    float c12x = by * cz - bz * cy;
    float c12y = bz * cx - bx * cz;
    float c12z = bx * cy - by * cx;
    acc = fmaf(ax, c12x, fmaf(ay, c12y, fmaf(az, c12z, acc)));
  }
  #pragma unroll
  for (int off = 16; off > 0; off >>= 1) acc += __shfl_down(acc, off, 32);
  int lane = threadIdx.x & 31;
  int wave = threadIdx.x >> 5;
  if (lane == 0) red[wave] = acc;
  __syncthreads();
  if (wave == 0) {
    float w = (lane < (kBlock / 32)) ? red[lane] : 0.0f;
    #pragma unroll
    for (int off = 4; off > 0; off >>= 1) w += __shfl_down(w, off, 32);
    if (lane == 0) fatomic_add(vol, w);
  }
}

// ---------------- pass 2: per-face forces, scatter-add into F -----------
// Face indices staged through LDS via async copies, double-buffered.
__global__ __launch_bounds__(kBlock)
void force_kernel(const float* __restrict__ x, const int* __restrict__ faces,
                  const float* __restrict__ vol_raw, float* __restrict__ F) {
  __shared__ int fbuf[2][kTileInts];
  const int tid = threadIdx.x;

  const float vol = (*vol_raw) * (1.0f / 6.0f);
  const float p   = kBulk * (kVt - vol) / kVt;

  const int nTiles = kFaces / kBlock;  // 15625, exact
  const uint32_t ldsBase = (uint32_t)(uintptr_t)(&fbuf[0][0]);
  const uint32_t ldsOff0 = ldsBase + (uint32_t)tid * 4u;                 // buf 0
  const uint32_t ldsOff1 = ldsBase + (uint32_t)kTileBytes + (uint32_t)tid * 4u; // buf 1

  int tile = blockIdx.x;
  if (tile < nTiles) issue_tile(ldsOff0, faces, tile, tid);

  int cur = 0;
  for (; tile < nTiles; tile += gridDim.x) {
    const int nextTile = tile + gridDim.x;
    const bool hasNext = nextTile < nTiles;
    if (hasNext) issue_tile(cur ? ldsOff0 : ldsOff1, faces, nextTile, tid);

    // oldest 3 async ops (current tile) complete in order
    if (hasNext) wait_asynccnt<3>();
    else         wait_asynccnt<0>();
    __syncthreads();   // publish current tile across all 8 waves

    const int i0 = fbuf[cur][3 * tid + 0];
    const int i1 = fbuf[cur][3 * tid + 1];
    const int i2 = fbuf[cur][3 * tid + 2];

    float ax, ay, az, bx, by, bz, cx, cy, cz;
    load3(x, i0, ax, ay, az);
    load3(x, i1, bx, by, bz);
    load3(x, i2, cx, cy, cz);

    // edges
    float e1x = bx - ax, e1y = by - ay, e1z = bz - az;   // v1 - v0
    float e2x = cx - ax, e2y = cy - ay, e2z = cz - az;   // v2 - v0
    float dx  = cx - bx, dy  = cy - by, dz  = cz - bz;   // v2 - v1

    // n = cross(e1, e2); nhat = n / (|n| + eps)
    float nx = e1y * e2z - e1z * e2y;
    float ny = e1z * e2x - e1x * e2z;
    float nz = e1x * e2y - e1y * e2x;
    float nn = sqrtf(fmaf(nx, nx, fmaf(ny, ny, nz * nz)));
    float inv = 1.0f / (nn + kEps);
    float hx = nx * inv, hy = ny * inv, hz = nz * inv;

    // dA0 = 0.5*cross(nhat, v2-v1); dA1 = -0.5*cross(nhat, e2); dA2 = 0.5*cross(nhat, e1)
    float a0x = 0.5f * (hy * dz - hz * dy);
    float a0y = 0.5f * (hz * dx - hx * dz);
    float a0z = 0.5f * (hx * dy - hy * dx);
    float a1x = -0.5f * (hy * e2z - hz * e2y);
    float a1y = -0.5f * (hz * e2x - hx * e2z);
    float a1z = -0.5f * (hx * e2y - hy * e2x);
    float a2x = 0.5f * (hy * e1z - hz * e1y);
    float a2y = 0.5f * (hz * e1x - hx * e1z);
    float a2z = 0.5f * (hx * e1y - hy * e1x);

    // dV0 = cross(v1,v2)/6; dV1 = cross(v2,v0)/6; dV2 = cross(v0,v1)/6
    const float s = 1.0f / 6.0f;
    float v0x = s * (by * cz - bz * cy);
    float v0y = s * (bz * cx - bx * cz);
    float v0z = s * (bx * cy - by * cx);
    float v1x = s * (cy * az - cz * ay);
    float v1y = s * (cz * ax - cx * az);
    float v1z = s * (cx * ay - cy * ax);
    float v2x = s * (ay * bz - az * by);
    float v2y = s * (az * bx - ax * bz);
    float v2z = s * (ax * by - ay * bx);

    // f_i = -sigma * dA_i + p * dV_i
    fatomic_add(F + 3 * (long)i0 + 0, fmaf(p, v0x, -kSigma * a0x));
    fatomic_add(F + 3 * (long)i0 + 1, fmaf(p, v0y, -kSigma * a0y));
    fatomic_add(F + 3 * (long)i0 + 2, fmaf(p, v0z, -kSigma * a0z));
    fatomic_add(F + 3 * (long)i1 + 0, fmaf(p, v1x, -kSigma * a1x));
    fatomic_add(F + 3 * (long)i1 + 1, fmaf(p, v1y, -kSigma * a1y));
    fatomic_add(F + 3 * (long)i1 + 2, fmaf(p, v1z, -kSigma * a1z));
    fatomic_add(F + 3 * (long)i2 + 0, fmaf(p, v2x, -kSigma * a2x));
    fatomic_add(F + 3 * (long)i2 + 1, fmaf(p, v2y, -kSigma * a2y));
    fatomic_add(F + 3 * (long)i2 + 2, fmaf(p, v2z, -kSigma * a2z));

    __syncthreads();   // all waves done reading fbuf[cur] before it is reissued
    cur ^= 1;
  }
}

// ---------------- pass 3: x += dt * F; F = 0; vol = 0 -------------------
__global__ __launch_bounds__(kBlock)
void update_kernel(float* __restrict__ x, float* __restrict__ F,
                   float* __restrict__ vol) {
  int i = blockIdx.x * blockDim.x + threadIdx.x;
  if (i < 3 * kVerts) {
    x[i] = fmaf(kDt, F[i], x[i]);
    F[i] = 0.0f;
  }
  if (i == 0) *vol = 0.0f;
}

extern "C" void kernel_launch(void* const* d_in, const int* in_sizes, int n_in,
                              void* d_out, int out_size, void* d_ws, size_t ws_size,
                              hipStream_t stream) {
  const float* verts = (const float*)d_in[0];
  const int*   faces = (const int*)d_in[1];   // jnp "int64" -> int32 (x64 disabled)
  float* x   = (float*)d_out;                 // positions live in d_out
  float* F   = (float*)d_ws;                  // 3*kVerts floats
  float* vol = F + 3 * (size_t)kVerts;        // 1 float accumulator

  const int nElems   = 3 * kVerts;
  const int gElems   = (nElems + kBlock - 1) / kBlock;
  const int gVolBlks = 4096;                  // grid-stride reduction
  const int gForce   = 4096;                  // ~3.8 tiles/block, double-buffered

  init_kernel<<<gElems, kBlock, 0, stream>>>(verts, x, F, vol);
  for (int it = 0; it < kIters; ++it) {
    vol_kernel<<<gVolBlks, kBlock, 0, stream>>>(x, faces, vol);
    force_kernel<<<gForce, kBlock, 0, stream>>>(x, faces, vol, F);
    update_kernel<<<gElems, kBlock, 0, stream>>>(x, F, vol);
  }
}